// SoftExpertLayer_721554505935
// MI455X (gfx1250) — compile-verified
//
#include <hip/hip_runtime.h>
#include <hip/hip_bf16.h>
#include <math.h>

// ---------------------------------------------------------------------------
// Soft-MoE expert layer for MI455X (gfx1250, wave32).
//
// Math collapses (P independent of slot k, G == 1/k):
//   dispatch = softmax_tokens(router)        (BS x E)
//   t        = dispatch^T @ x                (16p x H)   <- v_wmma_f32_16x16x4_f32
//   mid_e    = silu(t_e @ up_w[e/2] + up_b)  (E x DFF)   <- split-K matvec, async-LDS stage
//   o_e      = mid_e @ down_w[e] + down_b    (E x H)     <- split-K matvec, async-LDS stage
//   y        = dispatch @ O                  (BS x H)    <- v_wmma_f32_16x16x4_f32
//
// Traffic: x 64MB + up_w 179MB + down_w 358MB + y 64MB ~ 665MB, <1 GFLOP
// -> pure bandwidth problem (floor ~29us @ 23.3 TB/s). Weight-streaming
// kernels are split along the reduction dim (atomic f32 accumulation) so
// enough workgroups exist to saturate HBM.
// ---------------------------------------------------------------------------

#define NUM_EXPERTS 8
#define NUM_GROUPS  4
#define HIDDEN      2048
#define D_FF        5461
#define MID_STRIDE  5472          // D_FF padded to multiple of 32
#define BSZ         8192          // BATCH * SEQ
#define MPAD        16            // experts padded to one WMMA tile
#define HCHUNK      256           // up-kernel reduction chunk (8 chunks)
#define FCHUNK      352           // down-kernel reduction chunk (16 chunks cover 5632)
#define FCHUNKS     16

typedef float v2f __attribute__((ext_vector_type(2)));
typedef float v8f __attribute__((ext_vector_type(8)));

#if defined(__HIP_DEVICE_COMPILE__)
#define ATOMIC_ADD_F32(p, v) unsafeAtomicAdd((p), (v))   // -> global_atomic_add_f32
#else
#define ATOMIC_ADD_F32(p, v) atomicAdd((p), (v))
#endif

// --- gfx1250 async global->LDS staging (ASYNCcnt path), guarded fallback ----
#if defined(__HIP_DEVICE_COMPILE__) && __has_builtin(__builtin_amdgcn_global_load_async_to_lds_b32)
typedef __attribute__((address_space(1))) int gas_int;   // global
typedef __attribute__((address_space(3))) int las_int;   // LDS
#define HAVE_ASYNC_LDS 1
#endif

__device__ __forceinline__ void stage_to_lds(float* lds_dst, const float* gsrc) {
#if defined(HAVE_ASYNC_LDS)
  __builtin_amdgcn_global_load_async_to_lds_b32(
      (gas_int*)gsrc, (las_int*)lds_dst, 0, 0);
#else
  *lds_dst = *gsrc;
#endif
}
__device__ __forceinline__ void stage_wait() {
#if defined(__HIP_DEVICE_COMPILE__) && __has_builtin(__builtin_amdgcn_s_wait_asynccnt)
  __builtin_amdgcn_s_wait_asynccnt(0);
#endif
  __syncthreads();
}

// ---------------------------------------------------------------------------
// 0a) zero dispT pad rows (8..15) + t accumulator (ws is poisoned by harness)
// ---------------------------------------------------------------------------
__global__ void moe_zero_kernel(float* __restrict__ p, int n) {
  int i = blockIdx.x * blockDim.x + threadIdx.x;
  if (i < n) p[i] = 0.0f;
}

// ---------------------------------------------------------------------------
// 0b) seed accumulators with the biases: midacc[e][f] = up_b[e/2][f],
//     Oacc[e][h] = down_b[e][h]. Split-K partials atomic-add on top.
// ---------------------------------------------------------------------------
__global__ void moe_bias_init_kernel(const float* __restrict__ up_b,
                                     const float* __restrict__ down_b,
                                     float* __restrict__ midacc,
                                     float* __restrict__ Oacc) {
  const int i = blockIdx.x * blockDim.x + threadIdx.x;
  if (i < NUM_EXPERTS * MID_STRIDE) {
    const int e = i / MID_STRIDE, f = i - e * MID_STRIDE;
    midacc[i] = (f < D_FF) ? up_b[(e >> 1) * D_FF + f] : 0.0f;
  }
  const int j = i - NUM_EXPERTS * MID_STRIDE;
  if (j >= 0 && j < NUM_EXPERTS * HIDDEN) Oacc[j] = down_b[j];
}

// ---------------------------------------------------------------------------
// 1) per-expert softmax over the 8192 tokens; writes dispatch TRANSPOSED
//    dispT[e*BSZ + b] so WMMA A-operand loads are lane-coalesced.
// ---------------------------------------------------------------------------
__global__ void moe_softmax_kernel(const float* __restrict__ rw,
                                   float* __restrict__ dispT) {
  __shared__ float red[256];
  const int e   = blockIdx.x;
  const int tid = threadIdx.x;

  float m = -INFINITY;
  for (int b = tid; b < BSZ; b += 256) m = fmaxf(m, rw[b * NUM_EXPERTS + e]);
  red[tid] = m; __syncthreads();
  for (int s = 128; s > 0; s >>= 1) {
    if (tid < s) red[tid] = fmaxf(red[tid], red[tid + s]);
    __syncthreads();
  }
  m = red[0]; __syncthreads();

  float sum = 0.0f;
  for (int b = tid; b < BSZ; b += 256) sum += __expf(rw[b * NUM_EXPERTS + e] - m);
  red[tid] = sum; __syncthreads();
  for (int s = 128; s > 0; s >>= 1) {
    if (tid < s) red[tid] += red[tid + s];
    __syncthreads();
  }
  const float inv = 1.0f / red[0];

  for (int b = tid; b < BSZ; b += 256)
    dispT[e * BSZ + b] = __expf(rw[b * NUM_EXPERTS + e] - m) * inv;
}

// ---------------------------------------------------------------------------
// 2) t(16 x 2048) = dispT(16 x 8192) @ x(8192 x 2048) via v_wmma_f32_16x16x4_f32
//    grid(128 n-tiles, 2) x 8 waves/block = 16 K-chunks of 512; partial tiles
//    accumulated into pre-zeroed t with global_atomic_add_f32.
// ---------------------------------------------------------------------------
__global__ void moe_gather_wmma_kernel(const float* __restrict__ x,
                                       const float* __restrict__ dispT,
                                       float* __restrict__ t) {
  const int lane  = threadIdx.x & 31;
  const int wave  = threadIdx.x >> 5;
  const int n0    = blockIdx.x * 16;
  const int k0    = (blockIdx.y * 8 + wave) * 512;
  const int r     = lane & 15;
  const int khalf = (lane >> 4) * 2;                   // lanes 16..31 hold K+2,K+3

  v8f c = {0.f, 0.f, 0.f, 0.f, 0.f, 0.f, 0.f, 0.f};
  const float* __restrict__ arow = dispT + (size_t)r * BSZ;

  for (int k = k0; k < k0 + 512; k += 4) {
    const int kb = k + khalf;
    v2f a, b;
    a.x = arow[kb];
    a.y = arow[kb + 1];
    b.x = x[(size_t)kb * HIDDEN + n0 + r];
    b.y = x[(size_t)(kb + 1) * HIDDEN + n0 + r];
    c = __builtin_amdgcn_wmma_f32_16x16x4_f32(false, a, false, b,
                                              (short)0, c, false, false);
  }

#pragma unroll
  for (int i = 0; i < 8; ++i) {
    const int m = i + (lane >> 4) * 8;
    ATOMIC_ADD_F32(&t[m * HIDDEN + n0 + r], c[i]);
  }
}

// ---------------------------------------------------------------------------
// 3) midacc[e] += t[e][h0:h0+256] @ up_w[g][h0:h0+256, :]  (both experts of a
//    group share one pass over up_w). grid(22 f-blk, 4 grp, 8 h-chunk) = 704.
// ---------------------------------------------------------------------------
__global__ void moe_up_partial_kernel(const float* __restrict__ up_w,
                                      const float* __restrict__ t,
                                      float* __restrict__ midacc) {
  __shared__ float t_lds[2 * HCHUNK];
  const int g  = blockIdx.y;
  const int h0 = blockIdx.z * HCHUNK;
  const int f  = blockIdx.x * 256 + threadIdx.x;

  for (int i = threadIdx.x; i < 2 * HCHUNK; i += 256) {
    const int er = i / HCHUNK, hh = i - er * HCHUNK;
    stage_to_lds(&t_lds[i], &t[(2 * g + er) * HIDDEN + h0 + hh]);
  }
  stage_wait();
  if (f >= D_FF) return;

  const float* __restrict__ w = up_w + (size_t)g * HIDDEN * D_FF
                                     + (size_t)h0 * D_FF + f;
  float a0 = 0.0f, a1 = 0.0f;
#pragma unroll 1
  for (int hh = 0; hh < HCHUNK; hh += 4) {
    __builtin_prefetch(w + (size_t)(hh + 32) * D_FF, 0, 0);  // global_prefetch_b8
#pragma unroll
    for (int j = 0; j < 4; ++j) {
      const float wv = w[(size_t)(hh + j) * D_FF];
      a0 = fmaf(wv, t_lds[hh + j], a0);
      a1 = fmaf(wv, t_lds[HCHUNK + hh + j], a1);
    }
  }
  ATOMIC_ADD_F32(&midacc[(2 * g + 0) * MID_STRIDE + f], a0);
  ATOMIC_ADD_F32(&midacc[(2 * g + 1) * MID_STRIDE + f], a1);
}

// ---------------------------------------------------------------------------
// 3b) mid = silu(midacc)  (bias was seeded into midacc; pad entries are 0)
// ---------------------------------------------------------------------------
__global__ void moe_silu_kernel(const float* __restrict__ midacc,
                                float* __restrict__ mid, int n) {
  const int i = blockIdx.x * blockDim.x + threadIdx.x;
  if (i < n) {
    const float z = midacc[i];
    mid[i] = z / (1.0f + __expf(-z));
  }
}

// ---------------------------------------------------------------------------
// 4) Oacc[e][h] += mid[e][f0:fend] @ down_w[e][f0:fend, h]
//    grid(8 h-blk, 8 experts, 16 f-chunks) = 1024 blocks; down_w (358 MB)
//    streamed coalesced exactly once.
// ---------------------------------------------------------------------------
__global__ void moe_down_partial_kernel(const float* __restrict__ down_w,
                                        const float* __restrict__ mid,
                                        float* __restrict__ Oacc) {
  __shared__ float m_lds[FCHUNK];
  const int e  = blockIdx.y;
  const int f0 = blockIdx.z * FCHUNK;
  const int h  = blockIdx.x * 256 + threadIdx.x;       // always < HIDDEN
  const int fend = (f0 + FCHUNK < D_FF) ? (f0 + FCHUNK) : D_FF;

  for (int i = threadIdx.x; i < FCHUNK; i += 256) {
    int f = f0 + i; if (f >= D_FF) f = D_FF - 1;       // clamp (pad lanes unused)
    stage_to_lds(&m_lds[i], &mid[e * MID_STRIDE + f]);
  }
  stage_wait();

  const float* __restrict__ w = down_w + (size_t)e * D_FF * HIDDEN + h;
  float acc = 0.0f;
  int f = f0;
#pragma unroll 1
  for (; f + 4 <= fend; f += 4) {
    __builtin_prefetch(w + (size_t)(f + 32) * HIDDEN, 0, 0);
    acc = fmaf(w[(size_t)(f + 0) * HIDDEN], m_lds[f - f0 + 0], acc);
    acc = fmaf(w[(size_t)(f + 1) * HIDDEN], m_lds[f - f0 + 1], acc);
    acc = fmaf(w[(size_t)(f + 2) * HIDDEN], m_lds[f - f0 + 2], acc);
    acc = fmaf(w[(size_t)(f + 3) * HIDDEN], m_lds[f - f0 + 3], acc);
  }
  for (; f < fend; ++f) acc = fmaf(w[(size_t)f * HIDDEN], m_lds[f - f0], acc);

  ATOMIC_ADD_F32(&Oacc[e * HIDDEN + h], acc);
}

// ---------------------------------------------------------------------------
// 5) y(8192 x 2048) = dispatch(8192 x 8) @ Oacc(8 x 2048): two chained
//    v_wmma_f32_16x16x4_f32 per 16x16 tile; store-bandwidth bound (64 MB).
// ---------------------------------------------------------------------------
__global__ void moe_combine_wmma_kernel(const float* __restrict__ dispT,
                                        const float* __restrict__ O,
                                        float* __restrict__ y) {
  const int lane  = threadIdx.x & 31;
  const int wave  = threadIdx.x >> 5;
  const int n0    = blockIdx.x * 16;
  const int m0    = (blockIdx.y * 8 + wave) * 16;
  const int r     = lane & 15;
  const int khalf = (lane >> 4) * 2;
  const int gm    = m0 + r;
  const int col   = n0 + r;

  v2f a0, a1, b0, b1;
  a0.x = dispT[(khalf + 0) * BSZ + gm];
  a0.y = dispT[(khalf + 1) * BSZ + gm];
  a1.x = dispT[(khalf + 4) * BSZ + gm];
  a1.y = dispT[(khalf + 5) * BSZ + gm];
  b0.x = O[(khalf + 0) * HIDDEN + col];
  b0.y = O[(khalf + 1) * HIDDEN + col];
  b1.x = O[(khalf + 4) * HIDDEN + col];
  b1.y = O[(khalf + 5) * HIDDEN + col];

  v8f c = {0.f, 0.f, 0.f, 0.f, 0.f, 0.f, 0.f, 0.f};
  c = __builtin_amdgcn_wmma_f32_16x16x4_f32(false, a0, false, b0, (short)0, c, false, false);
  c = __builtin_amdgcn_wmma_f32_16x16x4_f32(false, a1, false, b1, (short)0, c, false, false);

#pragma unroll
  for (int i = 0; i < 8; ++i) {
    const int m = m0 + i + (lane >> 4) * 8;
    y[(size_t)m * HIDDEN + col] = c[i];
  }
}

// ---------------------------------------------------------------------------
extern "C" void kernel_launch(void* const* d_in, const int* in_sizes, int n_in,
                              void* d_out, int out_size, void* d_ws, size_t ws_size,
                              hipStream_t stream) {
  const float* x      = (const float*)d_in[0];
  const float* rw     = (const float*)d_in[1];
  const float* up_w   = (const float*)d_in[2];
  const float* up_b   = (const float*)d_in[3];
  const float* down_w = (const float*)d_in[4];
  const float* down_b = (const float*)d_in[5];
  float* y  = (float*)d_out;
  float* ws = (float*)d_ws;

  float* dispT  = ws;                                    // 16 * 8192
  float* t      = dispT + MPAD * BSZ;                    // 16 * 2048
  float* midacc = t + MPAD * HIDDEN;                     // 8 * 5472
  float* mid    = midacc + NUM_EXPERTS * MID_STRIDE;     // 8 * 5472
  float* Oacc   = mid + NUM_EXPERTS * MID_STRIDE;        // 8 * 2048

  // zero dispT rows 8..15 + the t accumulator (contiguous region)
  const int zn = (MPAD - NUM_EXPERTS) * BSZ + MPAD * HIDDEN;   // 98304
  moe_zero_kernel<<<dim3((zn + 255) / 256), dim3(256), 0, stream>>>(
      dispT + NUM_EXPERTS * BSZ, zn);

  // seed accumulators with biases
  const int bn = NUM_EXPERTS * MID_STRIDE + NUM_EXPERTS * HIDDEN;   // 60160
  moe_bias_init_kernel<<<dim3((bn + 255) / 256), dim3(256), 0, stream>>>(
      up_b, down_b, midacc, Oacc);

  moe_softmax_kernel<<<dim3(NUM_EXPERTS), dim3(256), 0, stream>>>(rw, dispT);

  moe_gather_wmma_kernel<<<dim3(HIDDEN / 16, 2), dim3(256), 0, stream>>>(x, dispT, t);

  moe_up_partial_kernel<<<dim3((D_FF + 255) / 256, NUM_GROUPS, HIDDEN / HCHUNK),
                          dim3(256), 0, stream>>>(up_w, t, midacc);

  const int sn = NUM_EXPERTS * MID_STRIDE;
  moe_silu_kernel<<<dim3((sn + 255) / 256), dim3(256), 0, stream>>>(midacc, mid, sn);

  moe_down_partial_kernel<<<dim3(HIDDEN / 256, NUM_EXPERTS, FCHUNKS),
                            dim3(256), 0, stream>>>(down_w, mid, Oacc);

  moe_combine_wmma_kernel<<<dim3(HIDDEN / 16, BSZ / 128), dim3(256), 0, stream>>>(
      dispT, Oacc, y);
}